// NeuralSemiLagrangian_17824114279056
// MI455X (gfx1250) — compile-verified
//
#include <hip/hip_runtime.h>
#include <math.h>

typedef __attribute__((ext_vector_type(2))) float v2f;
typedef __attribute__((ext_vector_type(8))) float v8f;

#define BB 2
#define CC 64
#define OO 128              // 2*C output channels
#define HH 180
#define WW 360
#define HP (HH + 2)
#define WP (WW + 2)
#define NPIX (HH * WW)      // 64800
#define TILE 16
#define TILES_PER_IMG (NPIX / TILE)  // 4050

__device__ __forceinline__ float cubic1(float x) {
    const float A = -0.75f;
    return ((A + 2.0f) * x - (A + 3.0f)) * x * x + 1.0f;
}
__device__ __forceinline__ float cubic2(float x) {
    const float A = -0.75f;
    return ((A * x - 5.0f * A) * x + 8.0f * A) * x - 4.0f * A;
}

// Value of geo_cyclic_pad(hidden,1)[b,c,hp,wp] without materializing the pad.
// img points at hidden[b][c] plane (H*W floats).
__device__ __forceinline__ float sample_pad(const float* __restrict__ img, int hp, int wp) {
    int w0 = wp - 1;                       // undo lon pad (cyclic)
    if (w0 < 0) w0 += WW;
    else if (w0 >= WW) w0 -= WW;
    int h;
    if (hp == 0) {                          // north pole row: row 0 rolled by W/2
        h = 0; w0 += WW / 2; if (w0 >= WW) w0 -= WW;
    } else if (hp == HH + 1) {              // south pole row: row H-1 rolled by W/2
        h = HH - 1; w0 += WW / 2; if (w0 >= WW) w0 -= WW;
    } else {
        h = hp - 1;
    }
    return img[h * WW + w0];
}

__global__ __launch_bounds__(256) void nsl_fused_kernel(
    const float* __restrict__ hidden,   // (B,C,H,W)
    const float* __restrict__ latg,     // (B,H,W)
    const float* __restrict__ lonG,     // (B,H,W)
    const float* __restrict__ Wv,       // (128,64) row-major
    const float* __restrict__ bv,       // (128,)
    const float* __restrict__ dtp,      // scalar
    float* __restrict__ out)            // (B,C,H,W)
{
    __shared__ float Wlds[OO * CC];     // 32 KB: full weight matrix
    __shared__ float hid[CC * TILE];    // 4 KB:  hid[c*16 + i]
    __shared__ float vel[OO * 17];      // 8.5 KB, stride 17 vs bank conflicts
    __shared__ float lonl[TILE];
    __shared__ float spl[TILE];         // sin(lat_p) per pixel (channel-invariant)
    __shared__ float cpl[TILE];         // cos(lat_p) per pixel

    const int t = threadIdx.x;
    const int blk = blockIdx.x;
    const int b = blk / TILES_PER_IMG;
    const int tile = blk - b * TILES_PER_IMG;
    const int p0 = tile * TILE;

    const float* __restrict__ hb = hidden + (size_t)b * CC * NPIX;

    // ---- stage W_vel (L2 resident) and the hidden tile into LDS ----
    for (int e = t; e < OO * CC; e += 256) Wlds[e] = Wv[e];
    for (int e = t; e < CC * TILE; e += 256) {
        int c = e >> 4, i = e & 15;
        hid[e] = hb[c * NPIX + p0 + i];
    }
    if (t < TILE) {
        float lat_p = latg[(size_t)b * NPIX + p0 + t];
        lonl[t] = lonG[(size_t)b * NPIX + p0 + t];
        spl[t] = __sinf(lat_p);
        cpl[t] = __cosf(lat_p);
    }
    __syncthreads();

    // ---- GEMM via V_WMMA_F32_16X16X4_F32: D[o][pix] += W[o][k] * hid[k][pix] ----
    const int lane = t & 31;
    const int wv   = t >> 5;            // wave id 0..7 -> o-tile
    const int m    = lane & 15;         // M row (A) / N col (B)
    const int kh   = (lane >> 4) << 1;  // K sub-offset 0 or 2 per 16x4 fp32 layout
    v8f acc = {};
    for (int k0 = 0; k0 < CC; k0 += 4) {
        v2f afrag, bfrag;
        const float* wrow = &Wlds[(wv * 16 + m) * CC + k0 + kh];
        afrag.x = wrow[0];
        afrag.y = wrow[1];
        bfrag.x = hid[(k0 + kh) * TILE + m];
        bfrag.y = hid[(k0 + kh + 1) * TILE + m];
        acc = __builtin_amdgcn_wmma_f32_16x16x4_f32(
            false, afrag, false, bfrag, (short)0, acc, false, false);
    }
    // C/D layout: VGPR r, lane l -> M = r + 8*(l>=16), N = l&15
    {
        const int mhi = (lane >> 4) * 8;
        for (int r = 0; r < 8; ++r) {
            int o = wv * 16 + r + mhi;
            vel[o * 17 + m] = acc[r] + bv[o];
        }
    }
    __syncthreads();

    // ---- per-point semi-Lagrangian departure + bicubic resample ----
    const float dt      = dtp[0];
    const float min_lat = latg[0];
    const float max_lat = latg[(size_t)(HH - 1) * WW];
    const float min_lon = lonG[0];
    const float max_lon = lonG[WW - 1];
    const float two_pi  = 6.2831853071795864f;
    const float inv2pi  = 0.15915494309189535f;
    const float sx = (float)WW / (float)WP;
    const float sy = (float)HH / (float)HP;
    const float glon = 2.0f / (max_lon - min_lon);
    const float glat = 2.0f / (max_lat - min_lat);

    for (int r = 0; r < 4; ++r) {
        int idx = t + 256 * r;          // 1024 (c,pixel) points per block
        int c = idx >> 4;
        int i = idx & 15;

        float u = vel[c * 17 + i];
        float v = vel[(CC + c) * 17 + i];
        float lon_p = lonl[i];
        float sp = spl[i];
        float cp = cpl[i];

        float lon_pr = -u * dt;
        float lat_pr = -v * dt;
        float slp = __sinf(lat_pr), clp = __cosf(lat_pr);
        float slo = __sinf(lon_pr), clo = __cosf(lon_pr);

        float sin_lat = slp * cp + clp * clo * sp;
        sin_lat = fminf(fmaxf(sin_lat, -1.0f + 1e-7f), 1.0f - 1e-7f);
        float lat = asinf(sin_lat);
        float num = clp * slo;
        float den = clp * clo * cp - slp * sp;
        float lon = lon_p + atan2f(num, den) + two_pi;
        lon = lon - floorf(lon * inv2pi) * two_pi;       // mod 2pi

        float gx = (lon - min_lon) * glon - 1.0f;
        float gy = (lat - min_lat) * glat - 1.0f;
        gx = (gx + 1.0f) - floorf((gx + 1.0f) * 0.5f) * 2.0f - 1.0f;  // mod 2 -> [-1,1)
        bool left  = gx <= 0.0f;
        bool outer = fabsf(gy) > 1.0f;
        if (outer) gx += left ? 1.0f : -1.0f;
        if (gy < -1.0f) gy = -(2.0f + gy);
        else if (gy > 1.0f) gy = 2.0f - gy;
        gx *= sx;
        gy *= sy;

        float ix = (gx + 1.0f) * 0.5f * (float)(WP - 1);
        float iy = (gy + 1.0f) * 0.5f * (float)(HP - 1);
        float ix0 = floorf(ix), iy0 = floorf(iy);
        float tx = ix - ix0,    ty = iy - iy0;

        float wxk[4] = { cubic2(tx + 1.0f), cubic1(tx), cubic1(1.0f - tx), cubic2(2.0f - tx) };
        float wyk[4] = { cubic2(ty + 1.0f), cubic1(ty), cubic1(1.0f - ty), cubic2(2.0f - ty) };

        int ixi = (int)ix0, iyi = (int)iy0;
        int xs[4], ys[4];
        for (int k = 0; k < 4; ++k) {
            int xx = ixi + k - 1; xs[k] = min(max(xx, 0), WP - 1);
            int yy = iyi + k - 1; ys[k] = min(max(yy, 0), HP - 1);
        }

        const float* __restrict__ img = hb + (size_t)c * NPIX;
        float accum = 0.0f;
        for (int j = 0; j < 4; ++j) {
            float rowv = 0.0f;
            for (int k = 0; k < 4; ++k)
                rowv += wxk[k] * sample_pad(img, ys[j], xs[k]);
            accum += wyk[j] * rowv;
        }

        out[((size_t)b * CC + c) * NPIX + p0 + i] = accum;
    }
}

extern "C" void kernel_launch(void* const* d_in, const int* in_sizes, int n_in,
                              void* d_out, int out_size, void* d_ws, size_t ws_size,
                              hipStream_t stream) {
    (void)in_sizes; (void)n_in; (void)out_size; (void)d_ws; (void)ws_size;
    const float* hidden = (const float*)d_in[0];
    const float* latg   = (const float*)d_in[1];
    const float* lonG   = (const float*)d_in[2];
    const float* Wv     = (const float*)d_in[3];
    const float* bv     = (const float*)d_in[4];
    const float* dtp    = (const float*)d_in[5];
    float* out = (float*)d_out;

    dim3 grid(BB * TILES_PER_IMG);   // 8100 blocks
    dim3 block(256);                 // 8 wave32
    nsl_fused_kernel<<<grid, block, 0, stream>>>(hidden, latg, lonG, Wv, bv, dtp, out);
}